// Voxelization_28063316312325
// MI455X (gfx1250) — compile-verified
//
#include <hip/hip_runtime.h>
#include <hip/hip_bf16.h>
#include <stdint.h>

#define GXD 1408
#define GYD 1600
#define GZD 40
#define MAX_POINTS 32
#define MAX_VOXELS 120000
#define HASH_BITS 22
#define HASH_SIZE (1u << HASH_BITS)
#define HASH_MASK (HASH_SIZE - 1u)
#define BLK 256

typedef __attribute__((address_space(3))) float lds_float_t;
typedef unsigned int v4u __attribute__((ext_vector_type(4)));
typedef int v4i __attribute__((ext_vector_type(4)));
typedef int v8i __attribute__((ext_vector_type(8)));

__device__ __forceinline__ uint32_t hash32(uint32_t x) {
    x *= 0xcc9e2d51u;
    x = (x << 15) | (x >> 17);
    x *= 0x1b873593u;
    x ^= x >> 16;
    return x & HASH_MASK;
}

__device__ __forceinline__ bool point_to_voxel(float px, float py, float pz,
                                               int& lin, int& cx, int& cy, int& cz) {
    cx = (int)floorf((px - 0.0f)   / 0.05f);
    cy = (int)floorf((py - -40.0f) / 0.05f);
    cz = (int)floorf((pz - -3.0f)  / 0.1f);
    if (cx < 0 || cx >= GXD || cy < 0 || cy >= GYD || cz < 0 || cz >= GZD) return false;
    lin = cz * (GYD * GXD) + cy * GXD + cx;
    return true;
}

// ---------- init kernels ----------
__global__ void k_zero_f(float* p, long n) {
    long i = (long)blockIdx.x * blockDim.x + threadIdx.x;
    long stride = (long)gridDim.x * blockDim.x;
    for (; i < n; i += stride) p[i] = 0.0f;
}

__global__ void k_init_ws(int* hkey, int* hmin, int* hslot, int* cnt, int* bcnt, int nblk) {
    long i = (long)blockIdx.x * blockDim.x + threadIdx.x;
    long stride = (long)gridDim.x * blockDim.x;
    for (; i < (long)HASH_SIZE; i += stride) {
        hkey[i]  = -1;
        hmin[i]  = 0x7fffffff;
        hslot[i] = MAX_VOXELS;
        if (i < MAX_VOXELS) cnt[i] = 0;
        if (i < nblk) bcnt[i] = 0;
    }
}

// ---------- phase 1: hash insert, record min point index per voxel ----------
// Each wave stages its 32 points (512B) into LDS with ONE Tensor Data Mover
// descriptor: 1-row tile, data_size=4B, tile_dim0 = #dwords (clamped at tail).
__global__ void k_insert(const float* __restrict__ pts, int N, int* hkey, int* hmin) {
    __shared__ float s_pts[BLK * 4];   // 4KB: 8 waves x 512B
    int tid  = threadIdx.x;
    int wave = __builtin_amdgcn_readfirstlane(tid >> 5);      // uniform per wave
    int waveBase = blockIdx.x * BLK + wave * 32;              // uniform
    int i = blockIdx.x * BLK + tid;
    int remain = N - waveBase;                                // uniform
    if (remain > 0) {
        int npts = remain < 32 ? remain : 32;
        uint32_t ndw = (uint32_t)npts * 4u;                   // dwords this wave copies
        uint32_t lds_base = (uint32_t)(uintptr_t)(lds_float_t*)&s_pts[0]
                          + (uint32_t)wave * 512u;
        uint64_t gaddr = (uint64_t)(uintptr_t)(pts + (size_t)waveBase * 4);
        // ---- D# group 0: count=1 | lds_addr | global_addr(57b) | type=2 ----
        v4u g0;
        g0.x = 1u;
        g0.y = lds_base;
        g0.z = (uint32_t)gaddr;
        g0.w = (uint32_t)((gaddr >> 32) & 0x01ffffffu) | (2u << 30);
        // ---- D# group 1: data_size=4B, 1 x ndw tile ----
        v8i g1;
        g1[0] = (int)(2u << 16);        // workgroup_mask=0, data_size=2 (4B)
        g1[1] = (int)(ndw << 16);       // tensor_dim0[15:0] in bits[63:48]
        g1[2] = (int)(1u << 16);        // tensor_dim0 hi=0 | tensor_dim1 lo=1
        g1[3] = (int)(ndw << 16);       // tensor_dim1 hi=0 | tile_dim0=ndw
        g1[4] = 1;                      // tile_dim1=1, tile_dim2=0
        g1[5] = (int)ndw;               // tensor_dim0_stride lo
        g1[6] = 0;                      // stride0 hi | stride1 lo
        g1[7] = 0;                      // stride1 hi
        v4i g2 = (v4i){0, 0, 0, 0};
        v4i g3 = (v4i){0, 0, 0, 0};
#if __clang_major__ >= 23
        __builtin_amdgcn_tensor_load_to_lds(g0, g1, g2, g3,
                                            (v8i){0, 0, 0, 0, 0, 0, 0, 0}, 0);
#else
        __builtin_amdgcn_tensor_load_to_lds(g0, g1, g2, g3, 0);
#endif
    }
    __builtin_amdgcn_s_wait_tensorcnt(0);
    if (i < N) {
        float px = s_pts[4 * tid + 0];
        float py = s_pts[4 * tid + 1];
        float pz = s_pts[4 * tid + 2];
        int lin, cx, cy, cz;
        if (point_to_voxel(px, py, pz, lin, cx, cy, cz)) {
            uint32_t h = hash32((uint32_t)lin);
            while (true) {
                int k = atomicCAS(&hkey[h], -1, lin);
                if (k == -1 || k == lin) { atomicMin(&hmin[h], i); break; }
                h = (h + 1) & HASH_MASK;
            }
        }
    }
}

// ---------- phase 2a: count leaders per block ----------
__global__ void k_leader_count(const float* __restrict__ pts, int N,
                               const int* __restrict__ hkey, const int* __restrict__ hmin,
                               int* bcnt) {
    __shared__ int red[BLK];
    int tid = threadIdx.x;
    int i = blockIdx.x * BLK + tid;
    int flag = 0;
    if (i < N) {
        float4 p = *(const float4*)(pts + (size_t)i * 4);
        int lin, cx, cy, cz;
        if (point_to_voxel(p.x, p.y, p.z, lin, cx, cy, cz)) {
            uint32_t h = hash32((uint32_t)lin);
            while (hkey[h] != lin) h = (h + 1) & HASH_MASK;
            flag = (hmin[h] == i) ? 1 : 0;
        }
    }
    red[tid] = flag;
    __syncthreads();
    for (int off = BLK / 2; off > 0; off >>= 1) {
        if (tid < off) red[tid] += red[tid + off];
        __syncthreads();
    }
    if (tid == 0) bcnt[blockIdx.x] = red[0];
}

// ---------- phase 2b: exclusive scan of block counts (single block) ----------
__global__ void k_scan(int* bcnt, int nblk, float* vnum_out) {
    __shared__ int s[BLK];
    int tid = threadIdx.x;
    int carry = 0;
    for (int base = 0; base < nblk; base += BLK) {
        int idx = base + tid;
        int v = (idx < nblk) ? bcnt[idx] : 0;
        s[tid] = v;
        __syncthreads();
        for (int off = 1; off < BLK; off <<= 1) {
            int t = (tid >= off) ? s[tid - off] : 0;
            __syncthreads();
            s[tid] += t;
            __syncthreads();
        }
        int incl  = s[tid];
        int total = s[BLK - 1];
        if (idx < nblk) bcnt[idx] = carry + (incl - v);   // exclusive offsets in place
        carry += total;
        __syncthreads();
    }
    if (tid == 0) {
        int vn = carry < MAX_VOXELS ? carry : MAX_VOXELS;
        vnum_out[0] = (float)vn;
    }
}

// ---------- phase 2c: assign slots (rank among leaders), write coors ----------
__global__ void k_assign(const float* __restrict__ pts, int N,
                         const int* __restrict__ hkey, const int* __restrict__ hmin,
                         int* hslot, const int* __restrict__ boff, float* coors) {
    __shared__ int s[BLK];
    int tid = threadIdx.x;
    int i = blockIdx.x * BLK + tid;
    int flag = 0;
    int lin = 0, cx = 0, cy = 0, cz = 0;
    uint32_t h = 0;
    bool leader = false;
    if (i < N) {
        float4 p = *(const float4*)(pts + (size_t)i * 4);
        if (point_to_voxel(p.x, p.y, p.z, lin, cx, cy, cz)) {
            h = hash32((uint32_t)lin);
            while (hkey[h] != lin) h = (h + 1) & HASH_MASK;
            leader = (hmin[h] == i);
            flag = leader ? 1 : 0;
        }
    }
    s[tid] = flag;
    __syncthreads();
    for (int off = 1; off < BLK; off <<= 1) {
        int t = (tid >= off) ? s[tid - off] : 0;
        __syncthreads();
        s[tid] += t;
        __syncthreads();
    }
    int excl = s[tid] - flag;
    if (leader) {
        int slot = boff[blockIdx.x] + excl;
        if (slot < MAX_VOXELS) {
            hslot[h] = slot;
            coors[(size_t)slot * 3 + 0] = (float)cz;
            coors[(size_t)slot * 3 + 1] = (float)cy;
            coors[(size_t)slot * 3 + 2] = (float)cx;
        }
    }
}

// ---------- phase 3: scatter point indices into per-slot lists ----------
// Stages points through LDS with the per-lane async copy path (ASYNCcnt).
__global__ void k_scatter(const float* __restrict__ pts, int N,
                          const int* __restrict__ hkey, const int* __restrict__ hslot,
                          int* cnt, int* idxbuf) {
    __shared__ float s_pts[BLK * 4];
    int tid = threadIdx.x;
    int i = blockIdx.x * BLK + tid;
    if (i < N) {
        uint32_t lds_off = (uint32_t)(uintptr_t)(lds_float_t*)&s_pts[4 * tid];
        uint64_t gaddr   = (uint64_t)(uintptr_t)(pts + (size_t)i * 4);
        asm volatile("global_load_async_to_lds_b128 %0, %1, off"
                     :: "v"(lds_off), "v"(gaddr) : "memory");
    }
    asm volatile("s_wait_asynccnt 0x0" ::: "memory");
    if (i < N) {
        float px = s_pts[4 * tid + 0];
        float py = s_pts[4 * tid + 1];
        float pz = s_pts[4 * tid + 2];
        int lin, cx, cy, cz;
        if (point_to_voxel(px, py, pz, lin, cx, cy, cz)) {
            uint32_t h = hash32((uint32_t)lin);
            while (hkey[h] != lin) h = (h + 1) & HASH_MASK;
            int slot = hslot[h];
            if (slot < MAX_VOXELS) {
                int p = atomicAdd(&cnt[slot], 1);
                if (p < MAX_POINTS) idxbuf[(size_t)slot * MAX_POINTS + p] = i;
            }
        }
    }
}

// ---------- phase 4: sort (restore original order), gather, write ----------
__global__ void k_write(const float* __restrict__ pts, const int* __restrict__ cnt,
                        const int* __restrict__ idxbuf, float* voxels, float* nump) {
    int slot = blockIdx.x * blockDim.x + threadIdx.x;
    if (slot >= MAX_VOXELS) return;
    int n = cnt[slot];
    if (n > MAX_POINTS) n = MAX_POINTS;
    int idx[MAX_POINTS];
    for (int r = 0; r < n; r++) idx[r] = idxbuf[(size_t)slot * MAX_POINTS + r];
    for (int r = 0; r < n; r++) __builtin_prefetch(pts + (size_t)idx[r] * 4, 0, 1);
    // insertion sort ascending: makes atomicAdd arrival order irrelevant (deterministic)
    for (int a = 1; a < n; a++) {
        int v = idx[a];
        int b = a - 1;
        while (b >= 0 && idx[b] > v) { idx[b + 1] = idx[b]; b--; }
        idx[b + 1] = v;
    }
    for (int r = 0; r < n; r++) {
        const float4 v = *(const float4*)(pts + (size_t)idx[r] * 4);
        *(float4*)(voxels + ((size_t)slot * MAX_POINTS + r) * 4) = v;
    }
    nump[slot] = (float)n;
}

extern "C" void kernel_launch(void* const* d_in, const int* in_sizes, int n_in,
                              void* d_out, int out_size, void* d_ws, size_t ws_size,
                              hipStream_t stream) {
    const float* pts = (const float*)d_in[0];
    int N = in_sizes[0] / 4;
    int nblk = (N + BLK - 1) / BLK;

    float* out    = (float*)d_out;
    float* voxels = out;                                              // 120000*32*4
    float* coors  = out + (size_t)MAX_VOXELS * MAX_POINTS * 4;        // 120000*3
    float* nump   = coors + (size_t)MAX_VOXELS * 3;                   // 120000
    float* vnum   = nump + MAX_VOXELS;                                // 1

    char* ws = (char*)d_ws;
    int* hkey   = (int*)ws;                      ws += (size_t)HASH_SIZE * 4;
    int* hmin   = (int*)ws;                      ws += (size_t)HASH_SIZE * 4;
    int* hslot  = (int*)ws;                      ws += (size_t)HASH_SIZE * 4;
    int* cnt    = (int*)ws;                      ws += (size_t)MAX_VOXELS * 4;
    int* idxbuf = (int*)ws;                      ws += (size_t)MAX_VOXELS * MAX_POINTS * 4;
    int* bcnt   = (int*)ws;                      ws += ((size_t)nblk * 4 + 255) & ~(size_t)255;

    k_zero_f<<<2048, 256, 0, stream>>>(out, (long)out_size);
    k_init_ws<<<4096, 256, 0, stream>>>(hkey, hmin, hslot, cnt, bcnt, nblk);
    k_insert<<<nblk, BLK, 0, stream>>>(pts, N, hkey, hmin);
    k_leader_count<<<nblk, BLK, 0, stream>>>(pts, N, hkey, hmin, bcnt);
    k_scan<<<1, BLK, 0, stream>>>(bcnt, nblk, vnum);
    k_assign<<<nblk, BLK, 0, stream>>>(pts, N, hkey, hmin, hslot, bcnt, coors);
    k_scatter<<<nblk, BLK, 0, stream>>>(pts, N, hkey, hslot, cnt, idxbuf);
    k_write<<<(MAX_VOXELS + 255) / 256, 256, 0, stream>>>(pts, cnt, idxbuf, voxels, nump);
}